// GNN_63934883168987
// MI455X (gfx1250) — compile-verified
//
#include <hip/hip_runtime.h>

#define N_NODES 100000
#define D_FEAT  128

typedef float v2f __attribute__((ext_vector_type(2)));
typedef float v8f __attribute__((ext_vector_type(8)));

// ---------------------------------------------------------------------------
// Scatter-add: agg[dst] += h[src] over E edges.
// One wave (32 lanes) per edge; lane l moves features [4l, 4l+3] (float4).
// Gather is a fully-coalesced 512B row read; adds are hardware
// global_atomic_add_f32 (no return) hitting L2 (x/agg both fit in 192MB L2).
// ---------------------------------------------------------------------------
__global__ __launch_bounds__(256) void gnn_scatter(
    const float* __restrict__ H,     // [N, 128] source features
    const int*   __restrict__ ei,    // [2, E] edge index (src row, dst row)
    float*       __restrict__ Agg,   // [N, 128] accumulator (pre-zeroed)
    int E)
{
    const int wave = threadIdx.x >> 5;
    const int lane = threadIdx.x & 31;
    const int e = blockIdx.x * 8 + wave;
    if (e >= E) return;

    const int src = ei[e];
    const int dst = ei[E + e];

    const float4 v = *(const float4*)(H + (size_t)src * D_FEAT + lane * 4);
    float* p = Agg + (size_t)dst * D_FEAT + lane * 4;
    unsafeAtomicAdd(p + 0, v.x);
    unsafeAtomicAdd(p + 1, v.y);
    unsafeAtomicAdd(p + 2, v.z);
    unsafeAtomicAdd(p + 3, v.w);
}

// ---------------------------------------------------------------------------
// Fused dual-GEMM + bias (+ReLU):
//   Out = act( Agg @ Wrel^T + H @ Wroot^T + b )
// Block = 256 threads = 8 waves, covers 16 rows x 128 cols.
// Each wave computes one 16x16 tile with V_WMMA_F32_16X16X4_F32 (exact fp32).
//
// Fragment layout per ISA (32-bit A 16x4 / B 4x16):
//   lanes 0-15 : M/N = lane,    VGPR0 = K+0, VGPR1 = K+1
//   lanes 16-31: M/N = lane-16, VGPR0 = K+2, VGPR1 = K+3
// B[k][j] = W[j][k]  ==>  B fragment loads use the same row-major float2
// pattern on W as A uses on the activation rows.
//
// Safe for Out aliasing H: every block reads only the 16 rows it writes,
// and __syncthreads() separates all reads from all writes.
// ---------------------------------------------------------------------------
__global__ __launch_bounds__(256) void gnn_gemm_fused(
    const float* __restrict__ Agg,    // [N, 128]  (rel input)
    const float* __restrict__ H,      // [N, 128]  (root input)
    const float* __restrict__ Wrel,   // [128, 128] row-major [j, k]
    const float* __restrict__ Wroot,  // [128, 128]
    const float* __restrict__ bias,   // [128]
    float*       __restrict__ Out,    // [N, 128]
    int do_relu)
{
    const int wave = threadIdx.x >> 5;     // 0..7 -> column tile
    const int lane = threadIdx.x & 31;
    const int m0   = blockIdx.x * 16;      // 16 rows per block
    const int n0   = wave * 16;
    const int lm   = lane & 15;
    const int kk   = (lane >> 4) * 2;      // 0 or 2

    const float* arow = Agg   + (size_t)(m0 + lm) * D_FEAT;
    const float* hrow = H     + (size_t)(m0 + lm) * D_FEAT;
    const float* wrel = Wrel  + (size_t)(n0 + lm) * D_FEAT;
    const float* wrt  = Wroot + (size_t)(n0 + lm) * D_FEAT;

    v8f acc = {};

    // Agg @ Wrel^T   (K = 128, 32 x K4 steps)
#pragma unroll
    for (int kc = 0; kc < 32; ++kc) {
        const int k = kc * 4 + kk;
        v2f a = *(const v2f*)(arow + k);
        v2f b = *(const v2f*)(wrel + k);
        acc = __builtin_amdgcn_wmma_f32_16x16x4_f32(
            false, a, false, b, (short)0, acc, false, false);
    }
    // + H @ Wroot^T
#pragma unroll
    for (int kc = 0; kc < 32; ++kc) {
        const int k = kc * 4 + kk;
        v2f a = *(const v2f*)(hrow + k);
        v2f b = *(const v2f*)(wrt + k);
        acc = __builtin_amdgcn_wmma_f32_16x16x4_f32(
            false, a, false, b, (short)0, acc, false, false);
    }

    // All H reads done before any Out writes (Out may alias H).
    __syncthreads();

    const float bj = bias[n0 + lm];
    const int mbase = m0 + ((lane >> 4) * 8);   // C/D layout: VGPR r -> row r (+8 for hi lanes)
#pragma unroll
    for (int r = 0; r < 8; ++r) {
        float v = acc[r] + bj;
        if (do_relu) v = fmaxf(v, 0.0f);
        Out[(size_t)(mbase + r) * D_FEAT + (n0 + lm)] = v;
    }
}

// ---------------------------------------------------------------------------
// Inputs (setup_inputs order):
//  0: x        [N,128] f32
//  1: edge_idx [2,E]   int
//  2: W1_rel   [128,128] f32
//  3: W1_root  [128,128] f32
//  4: b1       [128] f32
//  5: W2_rel   [128,128] f32
//  6: W2_root  [128,128] f32
//  7: b2       [128] f32
// Out: [N,128] f32.  ws: 51.2MB agg buffer.
// ---------------------------------------------------------------------------
extern "C" void kernel_launch(void* const* d_in, const int* in_sizes, int n_in,
                              void* d_out, int out_size, void* d_ws, size_t ws_size,
                              hipStream_t stream) {
    const float* x       = (const float*)d_in[0];
    const int*   ei      = (const int*)  d_in[1];
    const float* W1_rel  = (const float*)d_in[2];
    const float* W1_root = (const float*)d_in[3];
    const float* b1      = (const float*)d_in[4];
    const float* W2_rel  = (const float*)d_in[5];
    const float* W2_root = (const float*)d_in[6];
    const float* b2      = (const float*)d_in[7];

    const int E = in_sizes[1] / 2;
    const int N = in_sizes[0] / D_FEAT;   // 100000 (divisible by 16)

    float* agg = (float*)d_ws;            // [N,128]
    float* h   = (float*)d_out;           // layer-1 output lives in d_out

    const size_t agg_bytes = (size_t)N * D_FEAT * sizeof(float);
    const int scatter_blocks = (E + 7) / 8;
    const int gemm_blocks    = N / 16;

    // Layer 1: h = relu(segsum(x) @ W1_rel^T + x @ W1_root^T + b1)
    hipMemsetAsync(agg, 0, agg_bytes, stream);
    gnn_scatter<<<scatter_blocks, 256, 0, stream>>>(x, ei, agg, E);
    gnn_gemm_fused<<<gemm_blocks, 256, 0, stream>>>(agg, x, W1_rel, W1_root, b1, h, 1);

    // Layer 2: out = segsum(h) @ W2_rel^T + h @ W2_root^T + b2  (in place over h)
    hipMemsetAsync(agg, 0, agg_bytes, stream);
    gnn_scatter<<<scatter_blocks, 256, 0, stream>>>(h, ei, agg, E);
    gnn_gemm_fused<<<gemm_blocks, 256, 0, stream>>>(agg, h, W2_rel, W2_root, b2, (float*)d_out, 0);
}